// Linear_77421080478250
// MI455X (gfx1250) — compile-verified
//
#include <hip/hip_runtime.h>
#include <hip/hip_bf16.h>
#include <math.h>

// ---------------------------------------------------------------------------
// Segmented (block-diagonal) linear: y[n,v,i] = (1/sqrt(mul)) * sum_u x[n,u,i] W[u,v]
// per irrep block, as fp32 WMMA (V_WMMA_F32_16X16X4_F32) GEMMs.
//
// Round-1 fix: the irrep block is now a template parameter, so the per-wave
// v-tile loops are compile-time exact (512->4 tiles/wave, 256->2, 128->1).
// No runtime guards -> no s_and_saveexec around WMMA (EXEC provably all-1s),
// accumulators stay pinned in registers, K-loop body is straight-line.
//
// Memory strategy:
//   * x tile (16 rows x MUL) staged to LDS once per workgroup, reused by all
//     MUL/16 v-tiles -> x read from HBM exactly once.
//   * W (1.4 MB total) stays L2-resident; B fragments loaded per K-step.
//   * LDS is K-major with 17-float pitch: conflict-free stores and reads
//     (the two half-waves hit disjoint bank sets).
// ---------------------------------------------------------------------------

typedef __attribute__((ext_vector_type(2))) float v2f;
typedef __attribute__((ext_vector_type(8))) float v8f;

#define ROW_DIM   1920
#define LDS_PITCH 17

template<int MUL, int D, int NT>
__global__ __launch_bounds__(256) void seg_linear_wmma(
    const float* __restrict__ x,
    const float* __restrict__ w,
    float* __restrict__ y,
    int xoff, int woff, float scale)
{
    __shared__ float lds[MUL * LDS_PITCH];

    constexpr int SHIFT = (MUL == 512) ? 9 : (MUL == 256 ? 8 : 7);
    const int comp = blockIdx.y;                 // irrep component 0..D-1
    const int n0   = blockIdx.x << 4;            // 16-row tile base

    // ---- Stage A tile: 16 rows x MUL channels of component `comp` into LDS.
    // Consecutive threads walk consecutive channels u (coalesced for D=1,
    // short-stride for D=3/5). Trip count is a multiple of 256 -> no divergence.
    constexpr int TOTAL = MUL << 4;
    #pragma unroll 4
    for (int idx = threadIdx.x; idx < TOTAL; idx += 256) {
        const int u = idx & (MUL - 1);
        const int m = idx >> SHIFT;
        lds[u * LDS_PITCH + m] = x[(size_t)(n0 + m) * ROW_DIM + xoff + u * D + comp];
    }
    __syncthreads();

    // ---- Per-wave fragment coordinates (wave32).
    const int wave = threadIdx.x >> 5;           // 0..7
    const int lane = threadIdx.x & 31;
    const int half = lane >> 4;                  // 0: rows K+0/K+1, 1: rows K+2/K+3
    const int mrow = lane & 15;                  // A row / B column within the tile

    v8f acc[NT];
    #pragma unroll
    for (int t = 0; t < NT; ++t) acc[t] = (v8f)(0.0f);

    // Per-tile weight column base: tile t covers v in [ (wave + 8t)*16, +16 ).
    const float* __restrict__ wseg = w + woff;

    // ---- K loop: NT WMMAs per 4-deep K step; unroll x2 for load/WMMA overlap.
    #pragma unroll 2
    for (int k0 = 0; k0 < MUL; k0 += 4) {
        const int kh = k0 + 2 * half;
        v2f a;
        a.x = lds[(kh + 0) * LDS_PITCH + mrow];
        a.y = lds[(kh + 1) * LDS_PITCH + mrow];

        const float* wrow = wseg + (size_t)kh * MUL + mrow;
        #pragma unroll
        for (int t = 0; t < NT; ++t) {
            const int v0 = (wave + (t << 3)) << 4;
            v2f b;
            b.x = wrow[v0];
            b.y = wrow[v0 + MUL];
            acc[t] = __builtin_amdgcn_wmma_f32_16x16x4_f32(
                /*neg_a=*/false, a,
                /*neg_b=*/false, b,
                /*c_mod=*/(short)0, acc[t],
                /*reuse_a=*/false, /*reuse_b=*/false);
        }
    }

    // ---- Scale + store. D layout: VGPR r holds row (r + 8*half), col = mrow.
    #pragma unroll
    for (int t = 0; t < NT; ++t) {
        const int v = ((wave + (t << 3)) << 4) + mrow;
        const int colbase = xoff + v * D + comp;
        #pragma unroll
        for (int r = 0; r < 8; ++r) {
            const int n = n0 + r + 8 * half;
            y[(size_t)n * ROW_DIM + colbase] = acc[t][r] * scale;
        }
    }
}

extern "C" void kernel_launch(void* const* d_in, const int* in_sizes, int n_in,
                              void* d_out, int out_size, void* d_ws, size_t ws_size,
                              hipStream_t stream) {
    const float* x = (const float*)d_in[0];
    const float* w = (const float*)d_in[1];
    float* y = (float*)d_out;

    const int n_rows = in_sizes[0] / ROW_DIM;    // 16384
    const int rt = n_rows / 16;                  // 1024 row tiles

    // seg0: 512x0e  -> GEMM [N,512]x[512,512], 32 v-tiles = 4/wave
    seg_linear_wmma<512, 1, 4><<<dim3(rt, 1), 256, 0, stream>>>(
        x, w, y, 0, 0, 1.0f / sqrtf(512.0f));
    // seg1: 256x1o  -> 3 GEMMs [N,256]x[256,256], 16 v-tiles = 2/wave
    seg_linear_wmma<256, 3, 2><<<dim3(rt, 3), 256, 0, stream>>>(
        x, w, y, 512, 512 * 512, 1.0f / sqrtf(256.0f));
    // seg2: 128x2e  -> 5 GEMMs [N,128]x[128,128], 8 v-tiles = 1/wave
    seg_linear_wmma<128, 5, 1><<<dim3(rt, 5), 256, 0, stream>>>(
        x, w, y, 1280, 512 * 512 + 256 * 256, 1.0f / sqrtf(128.0f));
}